// ModelNew_14061722927792
// MI455X (gfx1250) — compile-verified
//
#include <hip/hip_runtime.h>

typedef float v2f __attribute__((ext_vector_type(2)));
typedef float v8f __attribute__((ext_vector_type(8)));

#define CIN   32
#define COUT  16
#define DSP   48            // input spatial
#define OE    49            // even-output spatial (acc[1..49])
#define OO    97            // final output spatial
#define XCH   (48*48*48)    // per-channel x stride (floats)

// ---------------------------------------------------------------------------
// Zero-fill the 97^3 output grid with 128-bit stores (out_size divisible by 4)
// ---------------------------------------------------------------------------
__global__ __launch_bounds__(256) void fill_zero_kernel(float4* __restrict__ out, long n4) {
    long i = (long)blockIdx.x * 256 + threadIdx.x;
    if (i < n4) out[i] = make_float4(0.f, 0.f, 0.f, 0.f);
}

// ---------------------------------------------------------------------------
// Implicit-GEMM full-conv via V_WMMA_F32_16X16X4_F32.
//   M = 16 ix positions, N = 16 output channels, K-step = 4 features
//   features per (cc, tap): { x[cc], x^2[cc], x[cc+16], x^2[cc+16] }
// A fragment (16x4 f32): lane<16 -> K0,K1 ; lane>=16 -> K2,K3 => each lane
// loads one x value (channel cc + 16*(lane>>4)) and squares it.
// B fragment (4x16 f32): lane half selects K row => {lin, quad} b64 from LDS.
// Block = 256 threads = 8 waves; wave w handles row iy = iy0 + w; loops over
// 4 ix strips of 16 (49 = 3*16+1, last strip masked).
// Inner cc loop fully unrolled with indexed addressing so the channel stride
// folds into the global/DS instruction immediate offsets (no pointer bumps).
// ---------------------------------------------------------------------------
__global__ __launch_bounds__(256) void conv_wmma_kernel(
    const float* __restrict__ x,      // (4,32,48,48,48)
    const float* __restrict__ w,      // (32,16,3,3,3)
    const float* __restrict__ bias,   // (16)
    float* __restrict__ out)          // (4,16,97,97,97)
{
    // LDS weight table: [t=27][cc=16][g=2][o=16] x {lin, quad} = 27648 floats
    __shared__ float ldsW[27 * 16 * 2 * 16 * 2];

    const int tid = threadIdx.x;

    // ---- transform weights into LDS (fold bias and the 2*w^2 term) ----
    for (int idx = tid; idx < 13824; idx += 256) {
        int o  = idx & 15;
        int g  = (idx >> 4) & 1;
        int cc = (idx >> 5) & 15;
        int t  = idx >> 9;                 // 0..26
        int ch = cc + 16 * g;
        int jz = t / 9, jy = (t / 3) % 3, jx = t % 3;
        int kz = 2 - jz, ky = 2 - jy, kx = 2 - jx;   // d = e-k  =>  k = 2-j
        float wv  = w[(ch * 16 + o) * 27 + kz * 9 + ky * 3 + kx];
        ldsW[2 * idx]     = (bias[o] + 1.0f) * wv;   // linear term weight
        ldsW[2 * idx + 1] = 2.0f * wv * wv;          // quadratic term weight
    }
    __syncthreads();

    const int wave = tid >> 5;
    const int lane = tid & 31;
    const int m    = lane & 15;   // M row (ix offset) and also o for store
    const int g    = lane >> 4;   // channel group / K-half select

    // block -> (n, iz, iy-tile)
    int b   = blockIdx.x;
    int n   = b / (OE * 7);
    int r   = b - n * (OE * 7);
    int iz  = r / 7;
    int iyt = r - iz * 7;
    int iy  = iyt * 8 + wave;
    if (iy >= OE) return;                  // wave-uniform exit (EXEC stays full)

    // channel-group base pointer (channels 16*g .. 16*g+15 via imm offsets)
    const float* __restrict__ xg = x + ((long)n * CIN + 16 * g) * (long)XCH;
    const v2f*   __restrict__ wt = (const v2f*)ldsW;

    // per-wave z/y clamps (uniform)
    int  dzc[3], dyc[3];
    bool okz[3], oky[3];
    #pragma unroll
    for (int j = 0; j < 3; ++j) {
        int dz = iz - 1 + j;
        okz[j] = (unsigned)dz < (unsigned)DSP;
        dzc[j] = min(max(dz, 0), DSP - 1);
        int dy = iy - 1 + j;
        oky[j] = (unsigned)dy < (unsigned)DSP;
        dyc[j] = min(max(dy, 0), DSP - 1);
    }

    for (int x0 = 0; x0 < 64; x0 += 16) {  // 4 strips of 16 ix positions
        // per-lane x clamps for this strip (3 taps in x)
        int  dxc[3];
        bool okx[3];
        #pragma unroll
        for (int jx = 0; jx < 3; ++jx) {
            int dx = x0 - 1 + jx + m;
            okx[jx] = (unsigned)dx < (unsigned)DSP;
            dxc[jx] = min(max(dx, 0), DSP - 1);
        }

        v8f acc = {0.f, 0.f, 0.f, 0.f, 0.f, 0.f, 0.f, 0.f};

        for (int jz = 0; jz < 3; ++jz) {
            for (int jy = 0; jy < 3; ++jy) {
                int  rowoff = (dzc[jz] * DSP + dyc[jy]) * DSP;   // uniform
                bool okzy   = okz[jz] && oky[jy];
                #pragma unroll
                for (int jx = 0; jx < 3; ++jx) {
                    bool ok = okzy && okx[jx];
                    const float* __restrict__ xp = xg + rowoff + dxc[jx];
                    int t = jz * 9 + jy * 3 + jx;
                    const v2f* __restrict__ wp = wt + ((t * 32 + g) * 16 + m);

                    #pragma unroll
                    for (int cc = 0; cc < 16; ++cc) {
                        float xv = xp[cc * XCH];        // imm-offset global b32
                        xv = ok ? xv : 0.0f;            // v_cndmask, EXEC full
                        v2f A = {xv, xv * xv};          // {x, x^2} -> K pair
                        v2f B = wp[cc * 32];            // imm-offset ds b64
                        acc = __builtin_amdgcn_wmma_f32_16x16x4_f32(
                            /*neg_a=*/false, A, /*neg_b=*/false, B,
                            /*c_mod=*/(short)0, acc,
                            /*reuse_a=*/false, /*reuse_b=*/false);
                    }
                }
            }
        }

        // store D: lane = (o | g<<4); VGPR v -> ix = x0 + v + 8*g; even sites
        #pragma unroll
        for (int v = 0; v < 8; ++v) {
            int ix = x0 + v + 8 * g;
            if (ix < OE) {
                long oidx = ((((long)n * COUT + m) * OO + 2 * iz) * OO + 2 * iy) * OO + 2 * ix;
                out[oidx] = acc[v];
            }
        }
    }
}

// ---------------------------------------------------------------------------
extern "C" void kernel_launch(void* const* d_in, const int* in_sizes, int n_in,
                              void* d_out, int out_size, void* d_ws, size_t ws_size,
                              hipStream_t stream) {
    const float* x    = (const float*)d_in[0];
    const float* wgt  = (const float*)d_in[1];
    const float* bias = (const float*)d_in[2];
    float* out = (float*)d_out;

    long n4 = (long)out_size / 4;                 // 58,411,072 / 4
    int fill_blocks = (int)((n4 + 255) / 256);
    fill_zero_kernel<<<fill_blocks, 256, 0, stream>>>((float4*)out, n4);

    int main_blocks = 4 * OE * 7;                 // (n=4) x (iz=49) x (iy tiles=7)
    conv_wmma_kernel<<<main_blocks, 256, 0, stream>>>(x, wgt, bias, out);
}